// GateLayer_41059887349950
// MI455X (gfx1250) — compile-verified
//
#include <hip/hip_runtime.h>

// ---------------------------------------------------------------------------
// CRY gate on 2^24 statevector. control bit = elem bit 20, target = elem bit 11.
// HBM-bound: 128 MiB total traffic -> ~5.5us floor at 23.3 TB/s.
// Bulk path: float4 streaming (b128, non-temporal), exactly-once traffic.
// Small slice via V_WMMA_F32_16X16X4_F32 (gate in A, data in B).
// ---------------------------------------------------------------------------

typedef __attribute__((ext_vector_type(4))) float f4;
typedef __attribute__((ext_vector_type(2))) float v2f;
typedef __attribute__((ext_vector_type(8))) float v8f;

#define CTRL_BIT  20u   // element-space control bit (axis 3)
#define TGT_BIT   11u   // element-space target bit  (axis 12)
#define CTRL_VBIT 18u   // vec4-space control bit
#define TGT_VBIT   9u   // vec4-space target bit

static constexpr unsigned TOTAL_V4_PAIRS    = 1u << 20;  // vec4 pairs w/ control=1
static constexpr unsigned COPY_V4           = 1u << 21;  // vec4s w/ control=0
static constexpr unsigned WMMA_SCALAR_PAIRS = 1u << 15;  // slice done by WMMA
static constexpr unsigned WMMA_TILES        = WMMA_SCALAR_PAIRS / 16u; // 2048
static constexpr unsigned WMMA_V4_PAIRS     = WMMA_SCALAR_PAIRS / 4u;  // 8192

// ---------------------------------------------------------------------------
// Bulk streaming kernel: copy half + rotate (vec4-pair) half minus WMMA slice.
// ---------------------------------------------------------------------------
__global__ void cry_bulk(const f4* __restrict__ psi4, f4* __restrict__ out4,
                         const float* __restrict__ theta_p, unsigned total) {
    unsigned t = blockIdx.x * blockDim.x + threadIdx.x;
    if (t >= total) return;

    if (t < COPY_V4) {
        // insert control=0 at vec bit 18
        unsigned v   = t;
        unsigned vec = (v & ((1u << CTRL_VBIT) - 1u)) | ((v >> CTRL_VBIT) << (CTRL_VBIT + 1u));
        f4 x = __builtin_nontemporal_load(&psi4[vec]);
        __builtin_nontemporal_store(x, &out4[vec]);
    } else {
        // vec4-pair index, skipping the WMMA slice (pairs map 1:1, order-preserving)
        unsigned q   = (t - COPY_V4) + WMMA_V4_PAIRS;
        unsigned lo  = q & ((1u << TGT_VBIT) - 1u);                                  // vec bits 0..8
        unsigned mid = (q >> TGT_VBIT) & ((1u << (CTRL_VBIT - TGT_VBIT - 1u)) - 1u); // vec bits 10..17
        unsigned hi  = q >> (CTRL_VBIT - 1u);                                        // vec bits 19..21
        unsigned vec0 = lo | (mid << (TGT_VBIT + 1u)) | (1u << CTRL_VBIT) | (hi << (CTRL_VBIT + 1u));
        unsigned vec1 = vec0 | (1u << TGT_VBIT);

        float th = 0.5f * theta_p[0];
        float c  = __cosf(th), s = __sinf(th);

        f4 x = __builtin_nontemporal_load(&psi4[vec0]);
        f4 y = __builtin_nontemporal_load(&psi4[vec1]);
        f4 o0 = c * x - s * y;
        f4 o1 = s * x + c * y;
        __builtin_nontemporal_store(o0, &out4[vec0]);
        __builtin_nontemporal_store(o1, &out4[vec1]);
    }
}

// ---------------------------------------------------------------------------
// WMMA kernel: D(16x16) = A(16x4) x B(4x16), f32.
// B: v0 = x in ALL 32 lanes, v1 = y in ALL 32 lanes (lanes 16-31 duplicate the
//    same addresses as lanes 0-15 -> unconditional, branch-free loads).
// Every plausible f32 B layout maps v0-low-lanes -> row K0 and v1-high-lanes
// -> row K3; the ambiguous middle rows (K1/K2) hold duplicated x or y, so the
// gate lives only in columns K0 and K3 of A:
//   A row0 = [c, 0, 0, -s],  row1 = [s, 0, 0, c],  rows 2..15 = 0.
// A layout (documented): lanes=M; v0 = K0|K2 (low|high), v1 = K1|K3.
//   -> a0: lane0=c, lane1=s;  a1: lane16=-s, lane17=c;  else 0.
// D: VGPR0 lanes0-15 = c*x - s*y, VGPR1 lanes0-15 = s*x + c*y.
// ---------------------------------------------------------------------------
__global__ void cry_wmma(const float* __restrict__ psi, float* __restrict__ out,
                         const float* __restrict__ theta_p) {
    const unsigned lane   = threadIdx.x & 31u;
    const unsigned wid    = blockIdx.x * (blockDim.x >> 5) + (threadIdx.x >> 5);
    const unsigned nwaves = gridDim.x * (blockDim.x >> 5);

    float th = 0.5f * theta_p[0];
    float c  = __cosf(th), s = __sinf(th);

    // Gate matrix in A operand (per-lane values, lanes = M rows).
    float a0 = (lane == 0u)  ? c  : (lane == 1u)  ? s : 0.0f;  // v0: K0 | K2
    float a1 = (lane == 16u) ? -s : (lane == 17u) ? c : 0.0f;  // v1: K1 | K3
    v2f A;  A.x = a0;  A.y = a1;

    const bool act = lane < 16u;

    for (unsigned tile = wid; tile < WMMA_TILES; tile += nwaves) {
        unsigned p    = tile * 16u + (lane & 15u);  // scalar pair id
        unsigned idx0 = (p & 0x7FFu) | (((p >> 11) & 0xFFu) << 12) |
                        (1u << CTRL_BIT) | ((p >> 19) << 21);
        unsigned idx1 = idx0 + (1u << TGT_BIT);

        // Unconditional: lanes 16-31 load the same 64B lines as lanes 0-15.
        float x = psi[idx0];
        float y = psi[idx1];
        v2f B;  B.x = x;  B.y = y;

        v8f C = {};
        v8f D = __builtin_amdgcn_wmma_f32_16x16x4_f32(
            /*neg_a=*/false, A, /*neg_b=*/false, B,
            /*c_mod=*/(short)0, C, /*reuse_a=*/false, /*reuse_b=*/false);

        if (act) {
            out[idx0] = D[0];  // row M=0: c*x - s*y
            out[idx1] = D[1];  // row M=1: s*x + c*y
        }
    }
}

// ---------------------------------------------------------------------------
extern "C" void kernel_launch(void* const* d_in, const int* in_sizes, int n_in,
                              void* d_out, int out_size, void* d_ws, size_t ws_size,
                              hipStream_t stream) {
    const float* psi     = (const float*)d_in[0];
    const float* theta_p = (const float*)d_in[1];
    float*       out     = (float*)d_out;

    // WMMA slice: 2048 tiles, 512 waves (64 blocks x 256 thr), 4 tiles/wave.
    cry_wmma<<<64, 256, 0, stream>>>(psi, out, theta_p);

    // Bulk: copies + remaining vec4 pairs.
    unsigned total  = COPY_V4 + (TOTAL_V4_PAIRS - WMMA_V4_PAIRS);
    unsigned blocks = (total + 255u) / 256u;
    cry_bulk<<<blocks, 256, 0, stream>>>((const f4*)psi, (f4*)out, theta_p, total);
}